// Encoder_8383776162326
// MI455X (gfx1250) — compile-verified
//
#include <hip/hip_runtime.h>
#include <hip/hip_fp16.h>
#include <stdint.h>

// Problem constants (from reference)
#define B_   256
#define F_   784
#define D_   1000
#define L_   100
#define DP   1024          // D padded to power of two
#define DT   256           // d-tile per block (128 half2 lanes)
#define BG   4             // batch rows per block
#define NBG  (B_ / BG)     // 64 batch groups
#define NDT  (DP / DT)     // 4 d-tiles

// Workspace layout
#define VALS_H_BYTES (L_ * DP * 2)      // 204800 B: half values [100][1024], pad=0
#define IDX4_OFF     VALS_H_BYTES
#define IDX4_BYTES   (NBG * F_ * 4)     // 200704 B: packed u8 levels, 4 rows/u32

// ---------------------------------------------------------------------------
// Prep: values_w (f32 +-1) -> half padded; x -> packed level indices (u8 x4)
// ---------------------------------------------------------------------------
__global__ void hdc_prep(const float* __restrict__ x,
                         const float* __restrict__ values_w,
                         __half* __restrict__ vals_h,
                         uint32_t* __restrict__ idx4)
{
    int tid = blockIdx.x * blockDim.x + threadIdx.x;
    if (tid < L_ * DP) {
        int l = tid >> 10;
        int d = tid & (DP - 1);
        float v = (d < D_) ? values_w[l * D_ + d] : 0.0f;   // pad with 0
        vals_h[tid] = __float2half_rn(v);                   // +-1 exact in f16
    } else {
        int t2 = tid - L_ * DP;
        if (t2 < NBG * F_) {
            int bg = t2 / F_;
            int f  = t2 - bg * F_;
            uint32_t w = 0;
            #pragma unroll
            for (int j = 0; j < BG; ++j) {
                float xx = x[(bg * BG + j) * F_ + f];
                int lvl = (int)rintf(xx * (float)(L_ - 1));  // RNE == jnp.round
                lvl = lvl < 0 ? 0 : (lvl > (L_ - 1) ? (L_ - 1) : lvl);
                w |= ((uint32_t)lvl) << (8 * j);
            }
            idx4[bg * F_ + f] = w;
        }
    }
}

// ---------------------------------------------------------------------------
// Encode: block = one 256-wide d-tile x 4 batch rows; 128 threads, 2 d's each.
// Value tile + packed index row staged to LDS via CDNA5 async global->LDS DMA.
// Exact f16 math: all products +-1, partial sums <= 784 < 2048.
// ---------------------------------------------------------------------------
__global__ __launch_bounds__(128)
void hdc_encode(const float* __restrict__ keys_w,
                const __half* __restrict__ vals_h,
                const uint32_t* __restrict__ idx4,
                float* __restrict__ out)
{
    __shared__ __align__(16) __half2   s_vals2[L_ * (DT / 2)];  // [100][128] h2 = 51200 B
    __shared__ __align__(16) uint32_t  s_idx[F_];               // 3136 B

    const int t  = threadIdx.x;     // 0..127: half2 lane within d-tile
    const int bg = blockIdx.y;      // 0..63
    const int d0 = blockIdx.x * DT; // 0,256,512,768

    // ---- async stage: value tile rows l=0..99, cols [d0, d0+256) ----------
    {
        const uint64_t gbase = (uint64_t)(uintptr_t)vals_h + (uint64_t)d0 * 2;
        const uint32_t lbase = (uint32_t)(uintptr_t)&s_vals2[0];
        // 100 rows * 512 B = 3200 chunks of 16 B
        for (int c = t; c < (L_ * DT * 2) / 16; c += 128) {
            int row = c >> 5;
            int c16 = c & 31;
            uint64_t ga = gbase + (uint64_t)row * (DP * 2) + (uint64_t)(c16 * 16);
            uint32_t la = lbase + (uint32_t)(row * (DT * 2) + c16 * 16);
            asm volatile("global_load_async_to_lds_b128 %0, %1, off"
                         :: "v"(la), "v"(ga) : "memory");
        }
        // packed index row: 784 u32 = 196 chunks of 16 B
        for (int c = t; c < (F_ * 4) / 16; c += 128) {
            uint64_t ga = (uint64_t)(uintptr_t)idx4
                        + (uint64_t)bg * (F_ * 4) + (uint64_t)(c * 16);
            uint32_t la = (uint32_t)(uintptr_t)&s_idx[0] + (uint32_t)(c * 16);
            asm volatile("global_load_async_to_lds_b128 %0, %1, off"
                         :: "v"(la), "v"(ga) : "memory");
        }
        asm volatile("s_wait_asynccnt 0x0" ::: "memory");
    }
    __syncthreads();

    const int d  = d0 + 2 * t;                      // even; this thread: d, d+1
    const int dk = (d <= D_ - 2) ? d : (D_ - 2);    // clamp OOB lanes to pair (998,999)

    __half2 acc0 = __float2half2_rn(0.0f);
    __half2 acc1 = __float2half2_rn(0.0f);
    __half2 acc2 = __float2half2_rn(0.0f);
    __half2 acc3 = __float2half2_rn(0.0f);

    const uint4* s_idxv = (const uint4*)s_idx;      // 196 x uint4

    for (int f4 = 0; f4 < F_ / 4; ++f4) {
        const uint4 ww = s_idxv[f4];                // ds_load_b128: 4 features' levels
        #pragma unroll
        for (int j = 0; j < 4; ++j) {
            const int f = 4 * f4 + j;
            const uint32_t w = (j == 0) ? ww.x : (j == 1) ? ww.y
                             : (j == 2) ? ww.z : ww.w;
            // keys pair (f32 +-1), converted once: v_cvt_pk_f16_f32
            const float2 kf = *(const float2*)(keys_w + (size_t)f * D_ + dk);
            const __half2 kh = __floats2half2_rn(kf.x, kf.y);
            // gathered value pairs from LDS: ds_load_b32 each, v_pk_fma_f16
            acc0 = __hfma2(kh, s_vals2[((w      ) & 0xFFu) * (DT / 2) + t], acc0);
            acc1 = __hfma2(kh, s_vals2[((w >>  8) & 0xFFu) * (DT / 2) + t], acc1);
            acc2 = __hfma2(kh, s_vals2[((w >> 16) & 0xFFu) * (DT / 2) + t], acc2);
            acc3 = __hfma2(kh, s_vals2[((w >> 24)        ) * (DT / 2) + t], acc3);
        }
    }

    if (d < D_) {                                   // d even, so d+1 <= 999 too
        const int b = bg * BG;
        const float2 a0 = __half22float2(acc0);
        const float2 a1 = __half22float2(acc1);
        const float2 a2 = __half22float2(acc2);
        const float2 a3 = __half22float2(acc3);
        out[(size_t)(b + 0) * D_ + d]     = (a0.x > 0.f) ? 1.f : -1.f;
        out[(size_t)(b + 0) * D_ + d + 1] = (a0.y > 0.f) ? 1.f : -1.f;
        out[(size_t)(b + 1) * D_ + d]     = (a1.x > 0.f) ? 1.f : -1.f;
        out[(size_t)(b + 1) * D_ + d + 1] = (a1.y > 0.f) ? 1.f : -1.f;
        out[(size_t)(b + 2) * D_ + d]     = (a2.x > 0.f) ? 1.f : -1.f;
        out[(size_t)(b + 2) * D_ + d + 1] = (a2.y > 0.f) ? 1.f : -1.f;
        out[(size_t)(b + 3) * D_ + d]     = (a3.x > 0.f) ? 1.f : -1.f;
        out[(size_t)(b + 3) * D_ + d + 1] = (a3.y > 0.f) ? 1.f : -1.f;
    }
}

// ---------------------------------------------------------------------------
extern "C" void kernel_launch(void* const* d_in, const int* in_sizes, int n_in,
                              void* d_out, int out_size, void* d_ws, size_t ws_size,
                              hipStream_t stream)
{
    const float* x        = (const float*)d_in[0];   // [256, 784]
    const float* keys_w   = (const float*)d_in[1];   // [784, 1000]
    const float* values_w = (const float*)d_in[2];   // [100, 1000]
    float* out = (float*)d_out;                      // [256, 1000]

    __half*   vals_h = (__half*)d_ws;
    uint32_t* idx4   = (uint32_t*)((char*)d_ws + IDX4_OFF);

    int total = L_ * DP + NBG * F_;                  // 152576 threads
    hdc_prep<<<(total + 255) / 256, 256, 0, stream>>>(x, values_w, vals_h, idx4);

    dim3 grid(NDT, NBG);                             // 4 x 64 = 256 blocks
    hdc_encode<<<grid, 128, 0, stream>>>(keys_w, vals_h, idx4, out);
}